// Conv1DTraining_31653908971662
// MI455X (gfx1250) — compile-verified
//
#include <hip/hip_runtime.h>
#include <math.h>

typedef __attribute__((ext_vector_type(2))) float v2f;
typedef __attribute__((ext_vector_type(8))) float v8f;

#define SEQ_LEN 4096
#define KSIZE   128
#define BATCH   8192
#define OUT_LEN (SEQ_LEN - KSIZE + 1)   // 3969

// workspace layout (float offsets)
#define WS_C     0        // 4096  : c[j] = kernel full-conv ones
#define WS_D     4096     // 8192  : d_conv per batch row
#define WS_TOT   12288    // 8192  : row totals
#define WS_KPRE  20480    // 129   : kernel prefix sums (Kpre[128] = Ksum)
#define WS_PART  24576    // 64*256: pass-2 per-block partials
// total: 40960 floats = 160 KB

// ---------------------------------------------------------------------------
// Kernel 0: build c[j] from the 128-tap kernel (tiny).
// c[j] = Kpre[j+1] (j<127) | Ksum (127<=j<=3968) | Ksum - Kpre[j-3968] (j>3968)
// ---------------------------------------------------------------------------
__global__ void k_prep(const float* __restrict__ kern, float* __restrict__ ws) {
    __shared__ float kpre[KSIZE + 1];
    const int t = threadIdx.x;
    if (t == 0) {
        float s = 0.f;
        kpre[0] = 0.f;
        for (int k = 0; k < KSIZE; ++k) { s += kern[k]; kpre[k + 1] = s; }
    }
    __syncthreads();
    const float ksum = kpre[KSIZE];
    for (int i = t; i <= KSIZE; i += blockDim.x) ws[WS_KPRE + i] = kpre[i];
    for (int j = t; j < SEQ_LEN; j += blockDim.x) {
        float cj;
        if (j < KSIZE - 1)         cj = kpre[j + 1];
        else if (j <= OUT_LEN - 1) cj = ksum;
        else                       cj = ksum - kpre[j - (OUT_LEN - 1)];
        ws[WS_C + j] = cj;
    }
}

// ---------------------------------------------------------------------------
// Kernel 1 (bandwidth-dominant, single read of x = 128 MiB):
// per 16-row tile compute pooled[b] = x_row . c and total[b] = sum(x_row)
// with V_WMMA_F32_16X16X4_F32.
//
// Since B (= c) is broadcast over all 16 N columns, only the A<->B K-slot
// *pairing* matters, not which column lands in which K slot. So per 8-column
// chunk each lane does ONE b128 load:
//   lanes 0-15  load x[row, k0   .. k0+3] -> K slots {0,1} of wmma #1/#2
//   lanes 16-31 load x[row, k0+4 .. k0+7] -> K slots {2,3} of wmma #1/#2
// and c is loaded with the identical float4 pattern so slots match:
//   wmma#1 consumes (av.xy | bv.xy), wmma#2 consumes (av.zw | bv.zw).
// B = ones (slot-independent) gives row totals from the same A tiles.
// D layout: lane 0 vgpr r = (M=r, N=0); lane 16 vgpr r = (M=8+r, N=0).
// ---------------------------------------------------------------------------
__global__ void __launch_bounds__(256) k_pass1(const float* __restrict__ x,
                                               const float* __restrict__ y,
                                               const float* __restrict__ bias,
                                               float* __restrict__ ws) {
    const int tid  = threadIdx.x;
    const int wave = tid >> 5;
    const int lane = tid & 31;
    const int half = lane >> 4;
    const int row_base = blockIdx.x * 16;
    const long long row = row_base + (lane & 15);

    const float* __restrict__ c    = ws + WS_C;
    const float* __restrict__ xrow = x + row * (long long)SEQ_LEN;

    v8f accP = {};            // dot with c
    v8f accT = {};            // row totals (dot with ones)
    v2f ones; ones[0] = 1.f; ones[1] = 1.f;

    const int kbeg = wave * (SEQ_LEN / 8);         // 512 columns per wave
    const int kend = kbeg + (SEQ_LEN / 8);
    for (int k0 = kbeg; k0 < kend; k0 += 8) {
        const int kk = k0 + (half << 2);           // half0: k0, half1: k0+4
        const float4 av = *(const float4*)(xrow + kk);
        const float4 bv = *(const float4*)(c + kk);
        v2f a1; a1[0] = av.x; a1[1] = av.y;
        v2f a2; a2[0] = av.z; a2[1] = av.w;
        v2f b1; b1[0] = bv.x; b1[1] = bv.y;
        v2f b2; b2[0] = bv.z; b2[1] = bv.w;
        accP = __builtin_amdgcn_wmma_f32_16x16x4_f32(false, a1, false, b1,
                                                     (short)0, accP, false, false);
        accT = __builtin_amdgcn_wmma_f32_16x16x4_f32(false, a1, false, ones,
                                                     (short)0, accT, false, false);
        accP = __builtin_amdgcn_wmma_f32_16x16x4_f32(false, a2, false, b2,
                                                     (short)0, accP, false, false);
        accT = __builtin_amdgcn_wmma_f32_16x16x4_f32(false, a2, false, ones,
                                                     (short)0, accT, false, false);
    }

    __shared__ float redP[8][16];
    __shared__ float redT[8][16];
    if (lane == 0) {
        for (int r = 0; r < 8; ++r) { redP[wave][r] = accP[r]; redT[wave][r] = accT[r]; }
    }
    if (lane == 16) {
        for (int r = 0; r < 8; ++r) { redP[wave][8 + r] = accP[r]; redT[wave][8 + r] = accT[r]; }
    }
    __syncthreads();

    if (tid < 16) {
        float pooled = 0.f, tot = 0.f;
        for (int w = 0; w < 8; ++w) { pooled += redP[w][tid]; tot += redT[w][tid]; }
        const int b = row_base + tid;
        // logits = (dot + OUT_LEN*bias)/OUT_LEN = dot/OUT_LEN + bias
        const float logits = pooled * (1.0f / OUT_LEN) + bias[0];
        const float sig = 1.0f / (1.0f + expf(-logits));
        ws[WS_D + b]   = (sig - y[b]) * (1.0f / OUT_LEN);   // d_conv
        ws[WS_TOT + b] = tot;
    }
}

// ---------------------------------------------------------------------------
// Kernel 2 (~8.3 MB): weighted column sums over the 254 boundary columns.
// thread t<127 : E[t] = sum_b D[b]*x[b,t]        (head cols 0..126)
// 127<=t<254   : F[t-127] = sum_b D[b]*x[b,3969+(t-127)] (tail cols)
// t==254       : G = sum_b D[b]*total[b];   t==255 : H = sum_b D[b]
// 64 blocks x 128 rows each; partials to ws (no atomics -> deterministic).
// ---------------------------------------------------------------------------
__global__ void __launch_bounds__(256) k_pass2(const float* __restrict__ x,
                                               float* __restrict__ ws) {
    const int t = threadIdx.x;
    const int rows_per_block = BATCH / 64;      // 128
    const int b0 = blockIdx.x * rows_per_block;
    const float* __restrict__ Dv = ws + WS_D;
    const float* __restrict__ Tv = ws + WS_TOT;

    int col = 0;
    if (t < 127)      col = t;
    else if (t < 254) col = OUT_LEN + (t - 127);   // 3969..4095

    float acc = 0.f;
    for (int i = 0; i < rows_per_block; ++i) {
        const int b = b0 + i;
        const float D = Dv[b];
        if (t < 254)       acc += D * x[(long long)b * SEQ_LEN + col];
        else if (t == 254) acc += D * Tv[b];
        else               acc += D;
    }
    ws[WS_PART + blockIdx.x * 256 + t] = acc;
}

// ---------------------------------------------------------------------------
// Kernel 3: reduce partials, build d_kernel via prefix/suffix scans, apply SGD.
// d_kernel[k] = (G - sum_{j<k} E[j] - sum_{t>=k} F[t]) / BATCH
// d_bias      = OUT_LEN * H / BATCH
// ---------------------------------------------------------------------------
__global__ void __launch_bounds__(256) k_final(const float* __restrict__ kern,
                                               const float* __restrict__ bias,
                                               float* __restrict__ out,
                                               const float* __restrict__ ws) {
    __shared__ float red[256];     // [0..126]=E, [127..253]=F, [254]=G, [255]=H
    const int t = threadIdx.x;
    float s = 0.f;
    for (int blk = 0; blk < 64; ++blk) s += ws[WS_PART + blk * 256 + t];
    red[t] = s;
    __syncthreads();

    if (t < KSIZE) {
        float pref = 0.f;
        for (int j = 0; j < t; ++j) pref += red[j];              // head prefix
        float suf = 0.f;
        for (int j = 127 + t; j < 254; ++j) suf += red[j];       // tail suffix
        const float G = red[254];
        const float dk = (G - pref - suf) * (1.0f / BATCH);
        out[t] = kern[t] - dk;                                   // LR = 1
    }
    if (t == 0) {
        const float db = ((float)OUT_LEN) * red[255] * (1.0f / BATCH);
        out[KSIZE] = bias[0] - db;
    }
}

extern "C" void kernel_launch(void* const* d_in, const int* in_sizes, int n_in,
                              void* d_out, int out_size, void* d_ws, size_t ws_size,
                              hipStream_t stream) {
    (void)in_sizes; (void)n_in; (void)out_size; (void)ws_size;
    const float* x    = (const float*)d_in[0];
    const float* y    = (const float*)d_in[1];
    const float* kern = (const float*)d_in[2];
    const float* bias = (const float*)d_in[3];
    float* out = (float*)d_out;
    float* ws  = (float*)d_ws;

    k_prep <<<1, 256, 0, stream>>>(kern, ws);
    k_pass1<<<BATCH / 16, 256, 0, stream>>>(x, y, bias, ws);
    k_pass2<<<64, 256, 0, stream>>>(x, ws);
    k_final<<<1, 256, 0, stream>>>(kern, bias, out, ws);
}